// BilateralFilter_14422500180435
// MI455X (gfx1250) — compile-verified
//
#include <hip/hip_runtime.h>
#include <stdint.h>

// Bilateral filter, MI455X (gfx1250).
// I: (4,1,480,640) f32, g: (1,49,480,640) f32 (spatially tiled -> 49 scalars),
// out: (4,480,640) f32.

#define KW      7
#define PAD     3
#define IMG_H   480
#define IMG_W   640
#define NBATCH  4
#define TILE_W  64
#define TILE_H  8
#define HALO_W  (TILE_W + 2 * PAD)   // 70
#define HALO_H  (TILE_H + 2 * PAD)   // 14
#define LDS_STRIDE 72                // padded row stride (floats)
// -(1/SIGMA_COLOR) * log2(e) with SIGMA_COLOR = 2*0.1^2 = 0.02
#define CEXP    (-72.134752044448169364f)

__global__ __launch_bounds__(TILE_W * TILE_H)
void bilateral_f32_kernel(const float* __restrict__ I,
                          const float* __restrict__ g,
                          float* __restrict__ out)
{
    __shared__ float tile[HALO_H * LDS_STRIDE];

    const int tx = threadIdx.x;
    const int ty = threadIdx.y;
    const int x0 = blockIdx.x * TILE_W;
    const int y0 = blockIdx.y * TILE_H;
    const int n  = blockIdx.z;

    const float* src = I + (size_t)n * (IMG_H * IMG_W);

    // ---- Stage tile + halo into LDS with async global->LDS DMA ------------
    // 980 elements loaded by 512 threads (<=2 per thread). Out-of-bounds
    // positions are zero-filled in LDS (matches reference zero padding).
    const int tid = ty * TILE_W + tx;
    for (int li = tid; li < HALO_H * HALO_W; li += TILE_W * TILE_H) {
        const int r  = li / HALO_W;
        const int c  = li - r * HALO_W;
        const int gy = y0 + r - PAD;
        const int gx = x0 + c - PAD;
        // Low 32 bits of a flat LDS address == byte offset in wave's LDS.
        const uint32_t lds_byte =
            (uint32_t)(uintptr_t)(&tile[r * LDS_STRIDE + c]);
        if ((unsigned)gy < (unsigned)IMG_H && (unsigned)gx < (unsigned)IMG_W) {
            const int goff = (gy * IMG_W + gx) * 4;
            // GVS form: lds_addr VGPR, 32-bit offset VGPR, 64-bit SGPR base.
            asm volatile("global_load_async_to_lds_b32 %0, %1, %2"
                         :
                         : "v"(lds_byte), "v"(goff), "s"(src)
                         : "memory");
        } else {
            tile[r * LDS_STRIDE + c] = 0.0f;
        }
    }
    asm volatile("s_wait_asynccnt 0" ::: "memory");
    __syncthreads();

    // ---- 49-tap bilateral accumulation from LDS ---------------------------
    const float Ic = tile[(ty + PAD) * LDS_STRIDE + (tx + PAD)];
    float num = 0.0f;
    float den = 0.0f;

#pragma unroll
    for (int dy = 0; dy < KW; ++dy) {
#pragma unroll
        for (int dx = 0; dx < KW; ++dx) {
            const float s = tile[(ty + dy) * LDS_STRIDE + (tx + dx)];
            const float d = s - Ic;
            // exp(-d^2/0.02) == exp2(d^2 * CEXP): raw v_exp_f32
            const float e = __builtin_amdgcn_exp2f(d * d * CEXP);
            // g is tiled over (H,W): tap k's weight lives at g[k*H*W].
            // Uniform address + constant offset -> scalar load, lives in SGPR.
            const float w = e * g[(size_t)(dy * KW + dx) * (IMG_H * IMG_W)];
            den += w;
            num = fmaf(w, s, num);
        }
    }

    // den >= 1 always (center tap contributes exp(0)*g_center == 1), so a
    // raw v_rcp_f32 (~1 ulp) is safe and much cheaper than the IEEE divide.
    const float inv_den = __builtin_amdgcn_rcpf(den);

    const int y = y0 + ty;
    const int x = x0 + tx;
    out[((size_t)n * IMG_H + y) * IMG_W + x] = num * inv_den;
}

extern "C" void kernel_launch(void* const* d_in, const int* in_sizes, int n_in,
                              void* d_out, int out_size, void* d_ws, size_t ws_size,
                              hipStream_t stream)
{
    (void)in_sizes; (void)n_in; (void)out_size; (void)d_ws; (void)ws_size;

    const float* I = (const float*)d_in[0];
    const float* g = (const float*)d_in[1];
    float* out     = (float*)d_out;

    dim3 block(TILE_W, TILE_H, 1);
    dim3 grid(IMG_W / TILE_W, IMG_H / TILE_H, NBATCH);
    bilateral_f32_kernel<<<grid, block, 0, stream>>>(I, g, out);
}